// OhemMSELoss_53584011985658
// MI455X (gfx1250) — compile-verified
//
#include <hip/hip_runtime.h>

// ---------------------------------------------------------------------------
// OHEM weighted-MSE top-k mean for MI455X (gfx1250, wave32).
// Pure bandwidth problem: ~804 MB streamed @ 23.3 TB/s => ~35 us floor.
// Exact top-k via 12-bit radix histogram + tie refinement.
// CDNA5 paths: v_wmma_f32_16x16x4_f32 wave reductions, NT loads, prefetch.
// ---------------------------------------------------------------------------

typedef float f4  __attribute__((ext_vector_type(4)));
typedef float v2f __attribute__((ext_vector_type(2)));
typedef float v8f __attribute__((ext_vector_type(8)));

#define K_KEPT 262144
#define NBINS  4096
#define BLOCK  256

struct Ctrl {
  int   candCount;
  int   bStar;
  int   cHi;
  int   need;
  float sumHi;
  int   pad[11];
};

// ---- 32-lane sum using one WMMA (A = partials 16x4, B = ones 4x16) --------
// Row sum r_m = v_m + v_{m+16}; D lane n vgpr j = r_{j(+8 for hi half)}.
// Sum of 8 D regs per lane = half-total; xor-16 shuffle completes it.
__device__ __forceinline__ float wave_reduce_sum(float v) {
#if __has_builtin(__builtin_amdgcn_wmma_f32_16x16x4_f32)
  v2f a; a[0] = v;    a[1] = 0.0f;
  v2f b; b[0] = 1.0f; b[1] = 1.0f;
  v8f c = {};
  v8f d = __builtin_amdgcn_wmma_f32_16x16x4_f32(
      /*neg_a=*/false, a, /*neg_b=*/false, b,
      /*c_mod=*/(short)0, c, /*reuse_a=*/false, /*reuse_b=*/false);
  float s = ((d[0] + d[1]) + (d[2] + d[3])) + ((d[4] + d[5]) + (d[6] + d[7]));
  s += __shfl_xor(s, 16, 32);
  return s;
#else
  for (int off = 16; off > 0; off >>= 1) v += __shfl_down(v, off, 32);
  return v;
#endif
}

// ---------------------------------------------------------------------------
__global__ void init_ws_kernel(int* hist, Ctrl* ctrl) {
  int i = blockIdx.x * blockDim.x + threadIdx.x;
  int stride = gridDim.x * blockDim.x;
  for (int b = i; b < NBINS; b += stride) hist[b] = 0;
  if (i == 0) {
    ctrl->candCount = 0;
    ctrl->bStar = 0;
    ctrl->cHi = 0;
    ctrl->need = K_KEPT;
    ctrl->sumHi = 0.0f;
  }
}

// ---- Pass 1: fused loss + 4096-bin histogram of top 12 float bits ---------
__global__ void loss_hist_kernel(const f4* __restrict__ p,
                                 const f4* __restrict__ t,
                                 const f4* __restrict__ w,
                                 int n4, float invNorm,
                                 int* __restrict__ hist) {
  __shared__ int h[NBINS];
  for (int b = threadIdx.x; b < NBINS; b += blockDim.x) h[b] = 0;
  __syncthreads();

  const int stride = gridDim.x * blockDim.x;
  for (int i = blockIdx.x * blockDim.x + threadIdx.x; i < n4; i += stride) {
    __builtin_prefetch((const void*)(p + i + stride), 0, 1);
    __builtin_prefetch((const void*)(t + i + stride), 0, 1);
    __builtin_prefetch((const void*)(w + i + stride), 0, 1);
    f4 pv = __builtin_nontemporal_load(p + i);
    f4 tv = __builtin_nontemporal_load(t + i);
    f4 wv = __builtin_nontemporal_load(w + i);
#pragma unroll
    for (int c = 0; c < 4; ++c) {
      float d = pv[c] - tv[c];
      float l = wv[c] * d * d * invNorm;           // l >= 0 => bit order == value order
      atomicAdd(&h[__float_as_uint(l) >> 20], 1);  // top 12 bits (sign always 0)
    }
  }
  __syncthreads();
  for (int b = threadIdx.x; b < NBINS; b += blockDim.x) {
    int c = h[b];
    if (c) atomicAdd(&hist[b], c);
  }
}

// ---- Pass 2: suffix scan over 4096 bins, single block ---------------------
__global__ void find_threshold_kernel(const int* __restrict__ hist,
                                      Ctrl* __restrict__ ctrl) {
  __shared__ int chunk[BLOCK];
  __shared__ int suffix[BLOCK];
  const int t = threadIdx.x;

  int s0 = 0;
  for (int b = 0; b < 16; ++b) s0 += hist[t * 16 + b];
  chunk[t] = s0;
  __syncthreads();

  if (t == 0) {
    int cum = 0;
    for (int c = BLOCK - 1; c >= 0; --c) { suffix[c] = cum; cum += chunk[c]; }
  }
  __syncthreads();

  const int above = suffix[t];
  if (above < K_KEPT && above + chunk[t] >= K_KEPT) {   // exactly one thread
    int cum = above;
    for (int b = t * 16 + 15; b >= t * 16; --b) {
      int c = hist[b];
      if (cum + c >= K_KEPT) {
        ctrl->bStar = b;
        ctrl->cHi   = cum;
        ctrl->need  = K_KEPT - cum;
        break;
      }
      cum += c;
    }
  }
}

// ---- Pass 3: sum strictly-above bin, compact ties -------------------------
__global__ void select_sum_kernel(const f4* __restrict__ p,
                                  const f4* __restrict__ t,
                                  const f4* __restrict__ w,
                                  int n4, float invNorm,
                                  Ctrl* __restrict__ ctrl,
                                  float* __restrict__ cand, int cap) {
  __shared__ float sAcc;
  if (threadIdx.x == 0) sAcc = 0.0f;
  __syncthreads();

  const unsigned keyStar = (unsigned)ctrl->bStar;
  float local = 0.0f;
  const int stride = gridDim.x * blockDim.x;
  for (int i = blockIdx.x * blockDim.x + threadIdx.x; i < n4; i += stride) {
    __builtin_prefetch((const void*)(p + i + stride), 0, 1);
    __builtin_prefetch((const void*)(t + i + stride), 0, 1);
    __builtin_prefetch((const void*)(w + i + stride), 0, 1);
    f4 pv = __builtin_nontemporal_load(p + i);
    f4 tv = __builtin_nontemporal_load(t + i);
    f4 wv = __builtin_nontemporal_load(w + i);
#pragma unroll
    for (int c = 0; c < 4; ++c) {
      float d = pv[c] - tv[c];
      float l = wv[c] * d * d * invNorm;
      unsigned key = __float_as_uint(l) >> 20;
      if (key > keyStar) {
        local += l;
      } else if (key == keyStar) {
        int idx = atomicAdd(&ctrl->candCount, 1);
        if (idx < cap) cand[idx] = l;
      }
    }
  }
  // Full-EXEC region: every thread of every full 8-wave block reaches here.
  float s = wave_reduce_sum(local);
  if ((threadIdx.x & 31) == 0) atomicAdd(&sAcc, s);
  __syncthreads();
  if (threadIdx.x == 0) atomicAdd(&ctrl->sumHi, sAcc);
}

// ---- Pass 4: exact refinement of remaining 20 bits over tie candidates ----
__global__ void final_select_kernel(Ctrl* __restrict__ ctrl,
                                    const float* __restrict__ cand, int cap,
                                    float* __restrict__ out, float kInv) {
  __shared__ int h[256];
  __shared__ unsigned sPref;
  __shared__ int sNeed;
  __shared__ float sAcc;
  const int tid = threadIdx.x;

  int cnt = ctrl->candCount;
  if (cnt > cap) cnt = cap;
  unsigned pref = ((unsigned)ctrl->bStar) << 20;
  unsigned mask = 0xFFF00000u;
  int need = ctrl->need;

  const int shifts[3] = {12, 4, 0};
  const int widths[3] = {8, 8, 4};
  for (int rnd = 0; rnd < 3; ++rnd) {
    const int sh = shifts[rnd];
    const int nb = 1 << widths[rnd];
    for (int b = tid; b < 256; b += blockDim.x) h[b] = 0;
    __syncthreads();
    for (int i = tid; i < cnt; i += blockDim.x) {
      unsigned u = __float_as_uint(cand[i]);
      if ((u & mask) == pref) atomicAdd(&h[(u >> sh) & (nb - 1)], 1);
    }
    __syncthreads();
    if (tid == 0) {
      sPref = pref;          // defaults in case counts never reach `need`
      sNeed = need;
      int cum = 0;
      for (int b = nb - 1; b >= 0; --b) {
        int c = h[b];
        if (cum + c >= need) {
          sPref = pref | ((unsigned)b << sh);
          sNeed = need - cum;
          break;
        }
        cum += c;
      }
    }
    __syncthreads();
    pref = sPref;
    need = sNeed;
    mask |= (unsigned)((nb - 1) << sh);
    __syncthreads();
  }

  // Sum candidates strictly greater than the exact threshold T = pref.
  float local = 0.0f;
  for (int i = tid; i < cnt; i += blockDim.x) {
    unsigned u = __float_as_uint(cand[i]);
    if (u > pref) local += cand[i];
  }
  if (tid == 0) sAcc = 0.0f;
  __syncthreads();
  float s = wave_reduce_sum(local);
  if ((tid & 31) == 0) atomicAdd(&sAcc, s);
  __syncthreads();
  if (tid == 0) {
    float total = ctrl->sumHi + sAcc + (float)need * __uint_as_float(pref);
    out[0] = total * kInv;
  }
}

// ---------------------------------------------------------------------------
extern "C" void kernel_launch(void* const* d_in, const int* in_sizes, int n_in,
                              void* d_out, int out_size, void* d_ws, size_t ws_size,
                              hipStream_t stream) {
  (void)n_in; (void)out_size;
  const float* p = (const float*)d_in[0];
  const float* t = (const float*)d_in[1];
  const float* w = (const float*)d_in[2];

  const int n  = in_sizes[0];          // 33,554,432
  const int n4 = n / 4;
  // norm_term = prod(spatial) * batch == total elements (channel dim == 1)
  const float invNorm = 1.0f / (float)n;

  int*  hist = (int*)d_ws;
  Ctrl* ctrl = (Ctrl*)((char*)d_ws + NBINS * sizeof(int));
  float* cand = (float*)((char*)d_ws + NBINS * sizeof(int) + sizeof(Ctrl));

  long capL = ((long)ws_size - (long)(NBINS * sizeof(int) + sizeof(Ctrl))) / 4;
  int cap = capL < 0 ? 0 : (capL > (1 << 24) ? (1 << 24) : (int)capL);

  const int blocks = 4096;             // 1M threads, 8 f4 per thread, grid-stride

  init_ws_kernel<<<16, BLOCK, 0, stream>>>(hist, ctrl);
  loss_hist_kernel<<<blocks, BLOCK, 0, stream>>>(
      (const f4*)p, (const f4*)t, (const f4*)w, n4, invNorm, hist);
  find_threshold_kernel<<<1, BLOCK, 0, stream>>>(hist, ctrl);
  select_sum_kernel<<<blocks, BLOCK, 0, stream>>>(
      (const f4*)p, (const f4*)t, (const f4*)w, n4, invNorm, ctrl, cand, cap);
  final_select_kernel<<<1, BLOCK, 0, stream>>>(
      ctrl, cand, cap, (float*)d_out, 1.0f / (float)K_KEPT);
}